// DistLoss_18949395710456
// MI455X (gfx1250) — compile-verified
//
#include <hip/hip_runtime.h>
#include <stdint.h>

typedef float v2f __attribute__((ext_vector_type(2)));
typedef float v8f __attribute__((ext_vector_type(8)));

#define IMG 256
#define NPIX 65536
#define BS 4
#define LPTS 128
#define NCHUNK 16
#define CHUNK_PIX 4096

// Workspace layout (bytes):
//   [0, 6144)          pooled: 512*3 f32
//   [8192, 532480)     cand:   512 pairs * 16 chunks * 8 u64
//   [532480, 565248)   tgt:    (4,128,8,2) f32
#define CAND_OFF 8192
#define TGT_OFF  (8192 + 512 * 128 * 8)

__device__ __forceinline__ void bubble_insert8(unsigned long long best[8],
                                               unsigned long long key) {
    // best[] sorted ascending; insert key, drop largest. Fully unrolled,
    // register-resident (no dynamic indexing -> no scratch spills).
    unsigned long long k = key;
#pragma unroll
    for (int q = 0; q < 8; ++q) {
        unsigned long long cur = best[q];
        unsigned long long mn = (k < cur) ? k : cur;
        unsigned long long mx = (k < cur) ? cur : k;
        best[q] = mn;
        k = mx;
    }
}

// ---------------- Kernel 1: pooled gather (with the reference's index scramble)
__global__ void k1_pool(const float* __restrict__ preds,
                        const float* __restrict__ ref,
                        float* __restrict__ pooled,
                        float* __restrict__ out) {
    int i = blockIdx.x * blockDim.x + threadIdx.x;
    if (i == 0) out[0] = 0.0f;           // zero the accumulator for k4
    if (i >= BS * LPTS) return;
    // flat pred row i -> coords
    float px = preds[i * 8 + 0];
    float py = preds[i * 8 + 1];
    float fx = rintf(px * 256.0f - 0.5f);   // round-half-even, matches jnp.round
    float fy = rintf(py * 256.0f - 0.5f);
    fx = fminf(fmaxf(fx, 0.0f), 255.0f);
    fy = fminf(fmaxf(fy, 0.0f), 255.0f);
    int ix = (int)fx, iy = (int)fy;
    int p = iy * IMG + ix;
    // reference: pooled[b, l] with i = l*bs + b  (the % bs scramble)
    int b = i & (BS - 1);
    int l = i >> 2;
#pragma unroll
    for (int c = 0; c < 3; ++c)
        pooled[(b * LPTS + l) * 3 + c] = ref[(size_t)(b * 3 + c) * NPIX + p];
}

// ---------------- Kernel 2: async-to-LDS staged top-8 scan over pixel chunks
__global__ void __launch_bounds__(256) k2_scan(const float* __restrict__ ref,
                                               const float* __restrict__ pooled,
                                               unsigned long long* __restrict__ cand) {
    __shared__ float smem[3 * CHUNK_PIX];   // 48 KB tile, SoA by channel
    int b = blockIdx.x >> 4;
    int chunk = blockIdx.x & 15;
    int tid = threadIdx.x;
    int chunkBase = chunk * CHUNK_PIX;

    // Stage 3 channel planes of this 4096-pixel chunk directly into LDS with
    // CDNA5 async copies (GLOBAL_LOAD_ASYNC_TO_LDS_B128, tracked by ASYNCcnt).
    // Flat pointers into the LDS aperture carry the workgroup-relative LDS
    // byte offset in addr[31:0] (ISA 10.2), which is exactly the VDST operand.
    {
        unsigned lds_base = (unsigned)(uintptr_t)(void*)smem;
        // 3 channels * 1024 float4 = 3072 b128 transfers; 12 per thread.
#pragma unroll
        for (int it = 0; it < 12; ++it) {
            int v = tid + it * 256;                 // [0, 3072)
            int c = v >> 10;                        // channel
            int e = v & 1023;                       // float4 index within plane
            const float* gp = ref + (size_t)(b * 3 + c) * NPIX + chunkBase + e * 4;
            unsigned long long ga = (unsigned long long)(uintptr_t)gp;
            unsigned lo = lds_base + (unsigned)(c * CHUNK_PIX + e * 4) * 4u;
            asm volatile("global_load_async_to_lds_b128 %0, %1, off"
                         :: "v"(lo), "v"(ga) : "memory");
        }
        asm volatile("s_wait_asynccnt 0x0" ::: "memory");
    }
    __syncthreads();

    // threads [0,128): half 0, threads [128,256): half 1  (h uniform per wave
    // -> all lanes broadcast-read the same LDS address: zero bank conflicts)
    int l = tid & 127;
    int h = tid >> 7;
    float p0 = pooled[(b * LPTS + l) * 3 + 0];
    float p1 = pooled[(b * LPTS + l) * 3 + 1];
    float p2 = pooled[(b * LPTS + l) * 3 + 2];

    unsigned long long best[8];
#pragma unroll
    for (int j = 0; j < 8; ++j) best[j] = ~0ull;

    int start = h * (CHUNK_PIX / 2);
    unsigned int gstart = (unsigned int)(chunkBase + start);
    for (int j = 0; j < CHUNK_PIX / 2; ++j) {
        int p = start + j;
        float sim = fabsf(smem[p] - p0) +
                    fabsf(smem[CHUNK_PIX + p] - p1) +
                    fabsf(smem[2 * CHUNK_PIX + p] - p2);
        unsigned long long key =
            ((unsigned long long)__float_as_uint(sim) << 32) | (gstart + j);
        if (key < best[7]) bubble_insert8(best, key);
    }

    // Merge the two halves of each l via LDS (reuse tile memory).
    __syncthreads();
    unsigned long long* sm64 = (unsigned long long*)smem;
#pragma unroll
    for (int j = 0; j < 8; ++j) sm64[tid * 8 + j] = best[j];
    __syncthreads();
    if (h == 0) {
        const unsigned long long* A = &sm64[tid * 8];
        const unsigned long long* B = &sm64[(tid + 128) * 8];
        size_t cb = ((size_t)(b * LPTS + l) * NCHUNK + chunk) * 8;
        int i0 = 0, i1 = 0;
#pragma unroll
        for (int j = 0; j < 8; ++j) {
            unsigned long long av = (i0 < 8) ? A[i0] : ~0ull;
            unsigned long long bv = (i1 < 8) ? B[i1] : ~0ull;
            if (av <= bv) { cand[cb + j] = av; ++i0; }
            else          { cand[cb + j] = bv; ++i1; }
        }
    }
}

// ---------------- Kernel 3: merge 16 chunk-candidates -> final top-8 -> targets
__global__ void k3_merge(const unsigned long long* __restrict__ cand,
                         float* __restrict__ tgt) {
    int i = blockIdx.x * blockDim.x + threadIdx.x;
    if (i >= BS * LPTS) return;
    const unsigned long long* c = cand + (size_t)i * (NCHUNK * 8);
    unsigned long long best[8];
#pragma unroll
    for (int j = 0; j < 8; ++j) best[j] = ~0ull;
    for (int j = 0; j < NCHUNK * 8; ++j) {
        unsigned long long key = c[j];
        if (key < best[7]) bubble_insert8(best, key);
    }
#pragma unroll
    for (int k = 0; k < 8; ++k) {
        unsigned int idx = (unsigned int)(best[k] & 0xFFFFFFFFull);
        float tx = (float)(idx & 255) * (1.0f / 256.0f);   // idx % 256 / 256
        float ty = (float)(idx >> 8) * (1.0f / 256.0f);    // idx / 256 / 256
        tgt[((size_t)i * 8 + k) * 2 + 0] = tx;
        tgt[((size_t)i * 8 + k) * 2 + 1] = ty;
    }
}

// ---------------- Kernel 4: WMMA distance matrix + min_K + mean
// dist^2(p,t) = [px, py, px^2+py^2, 1] . [-2tx, -2ty, 1, tx^2+ty^2]  (K=4)
// One wave per (b, 16-row l-tile); 8 column tiles of 16 targets (2 l's x 8 k).
__global__ void __launch_bounds__(32) k4_dist(const float* __restrict__ preds,
                                              const float* __restrict__ tgt,
                                              float* __restrict__ out) {
    __shared__ float sh[16 * 16];
    __shared__ float ssum[2];
    int b = blockIdx.x >> 3;
    int l0 = (blockIdx.x & 7) * 16;
    int lane = threadIdx.x;
    int row = lane & 15;

    // A 16x4 f32 layout: VGPR0 = K0 (lanes 0-15) / K2 (lanes 16-31), VGPR1 = K1/K3
    int lA = l0 + row;
    float px = preds[(size_t)(b * LPTS + lA) * 8 + 0];
    float py = preds[(size_t)(b * LPTS + lA) * 8 + 1];
    v2f a;
    a.x = (lane < 16) ? px : fmaf(px, px, py * py);   // K0 | K2
    a.y = (lane < 16) ? py : 1.0f;                    // K1 | K3

    float s0 = 0.0f, s1 = 0.0f;
    for (int t = 0; t < 8; ++t) {
        // B 4x16 f32 layout: VGPR0 = row0 (lanes 0-15) / row2 (16-31), VGPR1 = row1/row3
        int col = lane & 15;
        int lt = 2 * t + (col >> 3);          // which l within the tile pair
        int k = col & 7;
        int lsrc = (l0 + lt + LPTS - 1) & (LPTS - 1);   // roll(shift=1)
        float tx = tgt[((size_t)(b * LPTS + lsrc) * 8 + k) * 2 + 0];
        float ty = tgt[((size_t)(b * LPTS + lsrc) * 8 + k) * 2 + 1];
        v2f bb;
        bb.x = (lane < 16) ? (-2.0f * tx) : 1.0f;                  // row0 | row2
        bb.y = (lane < 16) ? (-2.0f * ty) : fmaf(tx, tx, ty * ty); // row1 | row3

        v8f cz = {0.f, 0.f, 0.f, 0.f, 0.f, 0.f, 0.f, 0.f};
        v8f d = __builtin_amdgcn_wmma_f32_16x16x4_f32(
            false, a, false, bb, (short)0, cz, false, false);

        // D layout: VGPR r holds row r (lanes 0-15) and row r+8 (lanes 16-31)
        int rbase = (lane < 16) ? 0 : 8;
#pragma unroll
        for (int r = 0; r < 8; ++r) sh[(rbase + r) * 16 + col] = d[r];
        __syncthreads();

        if (lane == 0) {   // l = l0+2t needs cols 0..7 of row 2t
            float m = sh[(2 * t) * 16 + 0];
#pragma unroll
            for (int c = 1; c < 8; ++c) m = fminf(m, sh[(2 * t) * 16 + c]);
            if ((l0 + 2 * t) != 0) s0 += sqrtf(fmaxf(m, 0.0f));  // skip l==0
        }
        if (lane == 1) {   // l = l0+2t+1 needs cols 8..15 of row 2t+1
            float m = sh[(2 * t + 1) * 16 + 8];
#pragma unroll
            for (int c = 9; c < 16; ++c) m = fminf(m, sh[(2 * t + 1) * 16 + c]);
            s1 += sqrtf(fmaxf(m, 0.0f));
        }
        __syncthreads();
    }
    if (lane == 0) ssum[0] = s0;
    if (lane == 1) ssum[1] = s1;
    __syncthreads();
    if (lane == 0)
        atomicAdd(out, (ssum[0] + ssum[1]) * (1.0f / (float)(BS * (LPTS - 1))));
}

extern "C" void kernel_launch(void* const* d_in, const int* in_sizes, int n_in,
                              void* d_out, int out_size, void* d_ws, size_t ws_size,
                              hipStream_t stream) {
    (void)in_sizes; (void)n_in; (void)out_size; (void)ws_size;
    const float* preds = (const float*)d_in[0];   // (4,128,8) f32
    const float* ref   = (const float*)d_in[1];   // (4,3,256,256) f32
    float* out = (float*)d_out;                   // scalar f32

    char* ws = (char*)d_ws;
    float* pooled = (float*)(ws);
    unsigned long long* cand = (unsigned long long*)(ws + CAND_OFF);
    float* tgt = (float*)(ws + TGT_OFF);

    k1_pool<<<2, 256, 0, stream>>>(preds, ref, pooled, out);
    k2_scan<<<BS * NCHUNK, 256, 0, stream>>>(ref, pooled, cand);
    k3_merge<<<2, 256, 0, stream>>>(cand, tgt);
    k4_dist<<<BS * 8, 32, 0, stream>>>(preds, tgt, out);
}